// AutoShape_5669356835376
// MI455X (gfx1250) — compile-verified
//
#include <hip/hip_runtime.h>
#include <stdint.h>

// ---- problem constants (match reference) ----
#define CONF_THRES 0.25f
#define IOU_THRES  0.45f
#define MAX_WH     7680.0f
#define BS      16
#define NA      25200
#define NO      85
#define NCLS    80
#define TOPK    2048
#define CAND    4096
#define MAXDET  300
#define NBIN    1024
#define BIN_BASE  0x3E800000u   // bits(0.25f); valid scores are strictly above
#define BIN_SHIFT 14            // 1024 bins over bits range [0.25, 1.0]

__device__ __forceinline__ uint32_t ballot32(bool p) {
#if __has_builtin(__builtin_amdgcn_ballot_w32)
  return __builtin_amdgcn_ballot_w32(p);
#else
  return (uint32_t)__ballot(p);
#endif
}

__device__ __forceinline__ uint32_t readlane32(uint32_t v, int lane) {
#if __has_builtin(__builtin_amdgcn_readlane)
  return (uint32_t)__builtin_amdgcn_readlane((int)v, lane);  // v_readlane_b32
#else
  return (uint32_t)__shfl((int)v, lane, 32);
#endif
}

// ---------------- K0: zero histograms ----------------
__global__ __launch_bounds__(256) void k0_zero(uint32_t* hist) {
  int t = blockIdx.x * 256 + threadIdx.x;
  if (t < BS * NBIN) hist[t] = 0u;
}

// ---------------- K1: decode + histogram ----------------
__global__ __launch_bounds__(256) void k1_decode(const float* __restrict__ pred,
                                                 float* __restrict__ score,
                                                 float4* __restrict__ box,
                                                 float* __restrict__ cls,
                                                 uint32_t* __restrict__ hist) {
  int t = blockIdx.x * 256 + threadIdx.x;
  if (t >= BS * NA) return;
  int img = t / NA;
  const float* p = pred + (size_t)t * NO;
  __builtin_prefetch(p + NO, 0, 0);  // global_prefetch_b8

  float x = p[0], y = p[1], w = p[2], h = p[3], obj = p[4];
  float best = -1.0f; int bi = 0;
#pragma unroll 8
  for (int c = 0; c < NCLS; ++c) {
    float cs = p[5 + c] * obj;            // cls_scores = pred[:,5:] * obj
    if (cs > best) { best = cs; bi = c; } // first-max == jnp.argmax tie rule
  }
  float hw = w * 0.5f, hh = h * 0.5f;
  box[t] = make_float4(x - hw, y - hh, x + hw, y + hh);
  cls[t] = (float)bi;

  bool valid = (obj > CONF_THRES) && (best > CONF_THRES);
  float s = valid ? best : -1.0f;
  score[t] = s;
  if (valid) {
    uint32_t bits = __float_as_uint(s);
    uint32_t bin = (bits - BIN_BASE) >> BIN_SHIFT;
    if (bin > NBIN - 1) bin = NBIN - 1;
    atomicAdd(&hist[img * NBIN + bin], 1u);
  }
}

// ---------------- K2: radix-select threshold via LDS suffix scan ----------------
__global__ __launch_bounds__(NBIN) void k2_thresh(const uint32_t* __restrict__ hist,
                                                  uint32_t* __restrict__ thr,
                                                  uint32_t* __restrict__ cnt) {
  __shared__ uint32_t s[NBIN];
  int img = blockIdx.x, b = threadIdx.x;
  s[b] = hist[img * NBIN + b];
  __syncthreads();
  for (int off = 1; off < NBIN; off <<= 1) {   // inclusive suffix sum, 10 steps
    uint32_t v = s[b];
    if (b + off < NBIN) v += s[b + off];
    __syncthreads();
    s[b] = v;
    __syncthreads();
  }
  uint32_t suf  = s[b];
  uint32_t sufn = (b + 1 < NBIN) ? s[b + 1] : 0u;
  if ((suf >= TOPK && sufn < TOPK) || (b == 0 && suf < TOPK))
    thr[img] = BIN_BASE + ((uint32_t)b << BIN_SHIFT);
  if (b == 0) cnt[img] = 0u;
}

// ---------------- K3: compact candidates above threshold ----------------
// key = (score_bits << 32) | ~index  -> descending u64 sort == stable top_k
__global__ __launch_bounds__(256) void k3_compact(const float* __restrict__ score,
                                                  const uint32_t* __restrict__ thr,
                                                  uint32_t* __restrict__ cnt,
                                                  unsigned long long* __restrict__ cand) {
  int t = blockIdx.x * 256 + threadIdx.x;
  if (t >= BS * NA) return;
  int img = t / NA, n = t % NA;
  float s = score[t];
  if (s > CONF_THRES) {
    uint32_t bits = __float_as_uint(s);
    if (bits >= thr[img]) {
      uint32_t pos = atomicAdd(&cnt[img], 1u);
      if (pos < CAND)
        cand[(size_t)img * CAND + pos] =
            ((unsigned long long)bits << 32) | (uint32_t)(~(uint32_t)n);
    }
  }
}

// ---------------- K4: per-image bitonic sort (descending) of 4096 keys ----------------
// Key staging: 32 KB contiguous global->LDS copy = ideal TDM (tensor_load_to_lds)
// tile. Descriptor: data_size=8B, tensor_dim0=tile_dim0=4096, higher dims 1.
typedef unsigned int uint32x4 __attribute__((ext_vector_type(4)));
typedef int          int32x4  __attribute__((ext_vector_type(4)));
typedef int          int32x8  __attribute__((ext_vector_type(8)));

__global__ __launch_bounds__(1024) void k4_sort(const unsigned long long* __restrict__ cand,
                                                const uint32_t* __restrict__ cnt,
                                                uint32_t* __restrict__ sidx,
                                                float* __restrict__ ssc) {
  __shared__ unsigned long long sk[CAND];  // 32 KB, only LDS object -> offset 0
  int img = blockIdx.x, tid = threadIdx.x;
  int n = (int)cnt[img]; if (n > CAND) n = CAND;

#if __has_builtin(__builtin_amdgcn_tensor_load_to_lds) && __has_builtin(__builtin_amdgcn_s_wait_tensorcnt)
  if (tid < 32) {  // one wave issues the DMA; TENSORcnt is per-wave
    uint64_t ga = (uint64_t)(uintptr_t)(cand + (size_t)img * CAND);
    uint32x4 g0 = { 1u,                      // count=1 valid descriptor
                    0u,                      // lds_addr = offset of sk == 0
                    (uint32_t)ga,            // global_addr[31:0]
                    (uint32_t)((ga >> 32) & 0x01FFFFFFu) | 0x80000000u }; // ga[56:32] | type=2
    int32x8 g1 = { 0x00030000,               // data_size=3 (8B), no mask/flags
                   0x10000000,               // tensor_dim0[15:0]=4096 in bits 79:48
                   0x00010000,               // tensor_dim1=1
                   0x10000000,               // tile_dim0=4096 (bits 127:112)
                   0,                        // tile_dim1=0 (unused), tile_dim2=0
                   0x00001000,               // tensor_dim0_stride=4096
                   0, 0 };
    int32x4 g2 = { 1, 1, 0, 0 };             // tensor_dim2=1, tensor_dim3=1
    int32x4 g3 = { 0, 0x00010000, 0, 0 };    // tensor_dim4=1
#if __clang_major__ >= 23
    int32x8 gz = { 0, 0, 0, 0, 0, 0, 0, 0 };
    __builtin_amdgcn_tensor_load_to_lds(g0, g1, g2, g3, gz, 0);
#else
    __builtin_amdgcn_tensor_load_to_lds(g0, g1, g2, g3, 0);
#endif
    __builtin_amdgcn_s_wait_tensorcnt(0);
  }
  __syncthreads();
  for (int i = tid; i < CAND; i += 1024)     // zero stale tail after DMA
    if (i >= n) sk[i] = 0ull;
  __syncthreads();
#else
  for (int i = tid; i < CAND; i += 1024)
    sk[i] = (i < n) ? cand[(size_t)img * CAND + i] : 0ull;
  __syncthreads();
#endif

  for (int kk = 2; kk <= CAND; kk <<= 1) {
    for (int jj = kk >> 1; jj > 0; jj >>= 1) {
      for (int i = tid; i < CAND; i += 1024) {
        int ixj = i ^ jj;
        if (ixj > i) {
          bool up = ((i & kk) == 0);
          unsigned long long a = sk[i], b = sk[ixj];
          if (up ? (a < b) : (a > b)) { sk[i] = b; sk[ixj] = a; }
        }
      }
      __syncthreads();
    }
  }
  for (int t = tid; t < TOPK; t += 1024) {
    unsigned long long k = sk[t];
    uint32_t idx = ~(uint32_t)k;
    uint32_t hi  = (uint32_t)(k >> 32);
    bool v = (hi != 0u) && (idx < NA);
    sidx[(size_t)img * TOPK + t] = v ? idx : 0xFFFFFFFFu;
    ssc[(size_t)img * TOPK + t]  = v ? __uint_as_float(hi) : -1.0f;
  }
}

// ---------------- K5: gather top-K boxes, add class offset, areas ----------------
__global__ __launch_bounds__(256) void k5_prep(const uint32_t* __restrict__ sidx,
                                               const float4* __restrict__ box,
                                               const float* __restrict__ cls,
                                               float4* __restrict__ rbox,
                                               float* __restrict__ scls,
                                               float4* __restrict__ ebox,
                                               float* __restrict__ area) {
  int t = blockIdx.x * 256 + threadIdx.x;
  if (t >= BS * TOPK) return;
  int img = t / TOPK;
  uint32_t idx = sidx[t];
  float4 b = make_float4(0.f, 0.f, 0.f, 0.f);
  float c = 0.f;
  if (idx < NA) {
    b = box[(size_t)img * NA + idx];
    c = cls[(size_t)img * NA + idx];
  }
  rbox[t] = b;
  scls[t] = c;
  float off = c * MAX_WH;
  float4 e = make_float4(b.x + off, b.y + off, b.z + off, b.w + off);
  ebox[t] = e;
  area[t] = (e.z - e.x) * (e.w - e.y);
}

// ---------------- K6: suppression bitmatrix (one wave32 ballot per 32 pairs) ----------------
__global__ __launch_bounds__(256) void k6_sup(const float4* __restrict__ ebox,
                                              const float* __restrict__ area,
                                              uint32_t* __restrict__ sup) {
  int gt = blockIdx.x * 256 + threadIdx.x;
  int gw = gt >> 5;
  int lane = threadIdx.x & 31;
  int img = gw / (TOPK * 64);
  int r   = gw % (TOPK * 64);
  int i   = r >> 6;
  int c   = r & 63;
  size_t row = ((size_t)img * TOPK + i) * 64;
  if (c * 32 + 31 <= i) {                 // entire chunk below diagonal: j>i false
    if (lane == 0) sup[row + c] = 0u;     // must still write (ws not re-poisoned)
    return;
  }
  int j = c * 32 + lane;
  size_t base = (size_t)img * TOPK;
  float4 a = ebox[base + i];
  float  ai = area[base + i];
  float4 b = ebox[base + j];
  float  aj = area[base + j];
  float iw = fmaxf(fminf(a.z, b.z) - fmaxf(a.x, b.x), 0.0f);
  float ih = fmaxf(fminf(a.w, b.w) - fmaxf(a.y, b.y), 0.0f);
  float inter = iw * ih;
  float iou = inter / (ai + aj - inter);
  bool p = (iou > IOU_THRES) && (j > i);  // NaN compares false
  uint32_t m = ballot32(p);
  if (lane == 0) sup[row + c] = m;
}

// ---------------- K7: sequential scan (1 wave/image) + emission ----------------
// keep bits live in 2 VGPRs/lane; 256B suppression rows double-buffered into LDS
// via async global->LDS (ASYNCcnt); keep-bit broadcast via v_readlane_b32.
__global__ __launch_bounds__(32) void k7_scan(const uint32_t* __restrict__ sup,
                                              const float* __restrict__ ssc,
                                              const float* __restrict__ rbox,
                                              const float* __restrict__ scls,
                                              float* __restrict__ out) {
  __shared__ uint32_t rowbuf[2][64];   // only LDS object -> offset 0
  int img = blockIdx.x;
  int lane = threadIdx.x;

  uint32_t k0 = 0, k1 = 0;             // keep0: sorted score > CONF_THRES
  const float* sc = ssc + (size_t)img * TOPK;
#pragma unroll 4
  for (int b = 0; b < 32; ++b) {
    if (sc[lane * 32 + b]        > CONF_THRES) k0 |= (1u << b);
    if (sc[(lane + 32) * 32 + b] > CONF_THRES) k1 |= (1u << b);
  }

  const uint32_t* rowbase = sup + (size_t)img * TOPK * 64;
  uint32_t goff = (uint32_t)(lane * 8);

  {  // prefetch row 0 into buffer 0 (each lane moves 8B; whole row = 256B)
    uint64_t gbase = (uint64_t)(uintptr_t)rowbase;
    uint32_t ldsoff = goff;
    asm volatile("global_load_async_to_lds_b64 %0, %1, %2"
                 :: "v"(ldsoff), "v"(goff), "s"(gbase) : "memory");
  }

  for (int i = 0; i < TOPK; ++i) {
    if (i + 1 < TOPK) {
      uint64_t gbase = (uint64_t)(uintptr_t)(rowbase + (size_t)(i + 1) * 64);
      uint32_t ldsoff = (uint32_t)(((i + 1) & 1) * 256) + goff;
      asm volatile("global_load_async_to_lds_b64 %0, %1, %2"
                   :: "v"(ldsoff), "v"(goff), "s"(gbase) : "memory");
      asm volatile("s_wait_asynccnt 0x1" ::: "memory");  // row i landed in LDS
    } else {
      asm volatile("s_wait_asynccnt 0x0" ::: "memory");
    }
    int word = i >> 5;
    uint32_t wv = readlane32((word < 32) ? k0 : k1, word & 31);
    if ((wv >> (i & 31)) & 1u) {        // keep[i] alive -> apply suppression row
      k0 &= ~rowbuf[i & 1][lane];
      k1 &= ~rowbuf[i & 1][lane + 32];
    }
  }

  // Survivors of the descending-sorted list are already in final top-MAX_DET
  // order (top_k over {sK | keep} is stable). Ballot-compact emission.
  const float* rb = rbox + (size_t)img * TOPK * 4;
  const float* cl = scls + (size_t)img * TOPK;
  float* o = out + (size_t)img * MAXDET * 6;
  int basec = 0;
  for (int w = 0; w < 64 && basec < MAXDET; ++w) {
    uint32_t m = readlane32((w < 32) ? k0 : k1, w & 31);
    if ((m >> lane) & 1u) {
      int pos = basec + __popc(m & ((1u << lane) - 1u));
      if (pos < MAXDET) {
        int j = w * 32 + lane;
        o[pos * 6 + 0] = rb[j * 4 + 0];
        o[pos * 6 + 1] = rb[j * 4 + 1];
        o[pos * 6 + 2] = rb[j * 4 + 2];
        o[pos * 6 + 3] = rb[j * 4 + 3];
        o[pos * 6 + 4] = sc[j];
        o[pos * 6 + 5] = cl[j];
      }
    }
    basec += __popc(m);
  }
  int start = basec < MAXDET ? basec : MAXDET;
  for (int r = start + lane; r < MAXDET; r += 32)
#pragma unroll
    for (int q = 0; q < 6; ++q) o[r * 6 + q] = 0.0f;
}

// ---------------- host launcher ----------------
extern "C" void kernel_launch(void* const* d_in, const int* in_sizes, int n_in,
                              void* d_out, int out_size, void* d_ws, size_t ws_size,
                              hipStream_t stream) {
  (void)in_sizes; (void)n_in; (void)out_size;
  const float* pred = (const float*)d_in[0];
  float* out = (float*)d_out;

  char* ws = (char*)d_ws;
  size_t off = 0;
  auto alloc = [&](size_t bytes) -> void* {
    void* p = ws + off;
    off = (off + bytes + 255) & ~(size_t)255;
    return p;
  };
  float*    score = (float*)alloc((size_t)BS * NA * 4);
  float4*   box   = (float4*)alloc((size_t)BS * NA * 16);
  float*    cls   = (float*)alloc((size_t)BS * NA * 4);
  uint32_t* hist  = (uint32_t*)alloc((size_t)BS * NBIN * 4);
  uint32_t* thr   = (uint32_t*)alloc((size_t)BS * 4);
  uint32_t* cnt   = (uint32_t*)alloc((size_t)BS * 4);
  unsigned long long* cand = (unsigned long long*)alloc((size_t)BS * CAND * 8);
  uint32_t* sidx  = (uint32_t*)alloc((size_t)BS * TOPK * 4);
  float*    ssc   = (float*)alloc((size_t)BS * TOPK * 4);
  float*    sclsA = (float*)alloc((size_t)BS * TOPK * 4);
  float4*   rbox  = (float4*)alloc((size_t)BS * TOPK * 16);
  float4*   ebox  = (float4*)alloc((size_t)BS * TOPK * 16);
  float*    area  = (float*)alloc((size_t)BS * TOPK * 4);
  uint32_t* sup   = (uint32_t*)alloc((size_t)BS * TOPK * 64 * 4);
  if (off > ws_size) return;  // ~20 MB required

  k0_zero   <<<(BS * NBIN + 255) / 256, 256, 0, stream>>>(hist);
  k1_decode <<<(BS * NA + 255) / 256, 256, 0, stream>>>(pred, score, box, cls, hist);
  k2_thresh <<<BS, NBIN, 0, stream>>>(hist, thr, cnt);
  k3_compact<<<(BS * NA + 255) / 256, 256, 0, stream>>>(score, thr, cnt, cand);
  k4_sort   <<<BS, 1024, 0, stream>>>(cand, cnt, sidx, ssc);
  k5_prep   <<<(BS * TOPK + 255) / 256, 256, 0, stream>>>(sidx, box, cls, rbox, sclsA, ebox, area);
  k6_sup    <<<BS * TOPK * 64 / 8, 256, 0, stream>>>(ebox, area, sup);
  k7_scan   <<<BS, 32, 0, stream>>>(sup, ssc, (const float*)rbox, sclsA, out);
}